// LSTM_37598143710127
// MI455X (gfx1250) — compile-verified
//
#include <hip/hip_runtime.h>
#include <cstdint>
#include <cstddef>

// LSTM scan: T=256 steps, B=128, F=1024.
// Per step: gates = [x_t | mask*h_{t-1}] @ [Wi; Wh] + b   (M=128, N=4096, K=2048)
// fused with gate math + state update. fp32 WMMA 16x16x4, K-pair-split LDS
// layout (no lane cndmasks), double-buffered async-to-LDS staging.

typedef float v2f __attribute__((ext_vector_type(2)));
typedef float v8f __attribute__((ext_vector_type(8)));

#define B_    128
#define F_    1024
#define F4_   4096
#define K_    2048
#define KT_   64        // K tile
#define NT_   128       // threads per block = 4 waves
#define NTILE 32        // K_/KT_

// A tile LDS layout: word = sel*A_SEL + kc*A_KC + row*2 + (k&1)
//   sel = ((k-k0)>>1)&1 selects {K0,K1} vs {K2,K3} half (bank phase +32)
#define A_SEL 2080
#define A_KC  130
#define ABUF  4160      // 2*A_SEL
// B tile LDS layout: word = g*B_G + sel*B_SEL + kc*32 + (k&1)*16 + n
#define B_SEL 544
#define B_G   1104
#define BBUF  4416      // 4*B_G

#if defined(__has_builtin)
#if __has_builtin(__builtin_amdgcn_global_load_async_to_lds_b64) && \
    __has_builtin(__builtin_amdgcn_global_load_async_to_lds_b128)
#define USE_ASYNC 1
#endif
#endif

#if defined(USE_ASYNC)
// Builtin signatures (from hipcc diagnostics): first param is a pointer to an
// N-int vector in the device (global, AS1) address space; LDS param is AS3.
typedef int vi2_t __attribute__((vector_size(8)));
typedef int vi4_t __attribute__((vector_size(16)));
typedef __attribute__((address_space(1))) vi2_t* g_v2_t;
typedef __attribute__((address_space(3))) vi2_t* l_v2_t;
typedef __attribute__((address_space(1))) vi4_t* g_v4_t;
typedef __attribute__((address_space(3))) vi4_t* l_v4_t;
#endif

__device__ __forceinline__ float sigmoidf_(float x) {
    return 1.0f / (1.0f + __expf(-x));
}

__device__ __forceinline__ void wait_async_le24() {
#if defined(USE_ASYNC)
#if __has_builtin(__builtin_amdgcn_s_wait_asynccnt)
    __builtin_amdgcn_s_wait_asynccnt(24);
#else
    asm volatile("s_wait_asynccnt 24" ::: "memory");
#endif
#endif
}
__device__ __forceinline__ void wait_async_0() {
#if defined(USE_ASYNC)
#if __has_builtin(__builtin_amdgcn_s_wait_asynccnt)
    __builtin_amdgcn_s_wait_asynccnt(0);
#else
    asm volatile("s_wait_asynccnt 0" ::: "memory");
#endif
#endif
}

// Block (jb, rb): columns [jb*16, jb*16+16) of each of the 4 gates,
//                 rows    [rb*64, rb*64+64).  Wave w: 16-row stripe.
__global__ __launch_bounds__(NT_)
void lstm_step_kernel(const float* __restrict__ x_t,
                      const int*   __restrict__ term_t,
                      const float* __restrict__ Wi,
                      const float* __restrict__ Wh,
                      const float* __restrict__ bias,
                      const float* __restrict__ h_in,
                      float*       __restrict__ h_out,
                      float*       __restrict__ c_state,
                      float*       __restrict__ y_t)
{
    __shared__ __align__(16) float As[2 * ABUF];   // double buffered
    __shared__ __align__(16) float Bs[2 * BBUF];

    const int tid  = threadIdx.x;
    const int lane = tid & 31;
    const int wv   = tid >> 5;
    const bool lo  = lane < 16;
    const int nl   = lane & 15;

    const int jb   = blockIdx.x;        // 0..63
    const int rb   = blockIdx.y;        // 0..1
    const int cb   = jb * 16;
    const int row0 = rb * 64;

    // termination mask for this lane's A-fragment row (applied to h side only)
    const float maskf = (term_t[row0 + wv * 16 + nl] != 0) ? 0.0f : 1.0f;

    // fragment base offsets (floats)
    const int aoff = (lo ? 0 : A_SEL) + (wv * 16 + nl) * 2;
    const int boff = (lo ? 0 : B_SEL) + nl;

    v8f acc0 = {}, acc1 = {}, acc2 = {}, acc3 = {};

    auto stage = [&](float* AsB, float* BsB, int k0) {
        const bool hside = (k0 >= F_);
        const float* Asrc = hside ? h_in : x_t;
        const float* Wp   = hside ? Wh : Wi;
        const int kb = hside ? (k0 - F_) : k0;
#if defined(USE_ASYNC)
        // A: 64 rows x 32 k-pairs -> 16 async b64 per thread
        #pragma unroll
        for (int i = 0; i < 16; ++i) {
            int id  = tid + i * NT_;
            int kp  = id & 31;            // k-pair index
            int row = id >> 5;            // 0..63
            const float* gp = Asrc + (size_t)(row0 + row) * F_ + kb + kp * 2;
            float* lp = AsB + (kp & 1) * A_SEL + (kp >> 1) * A_KC + row * 2;
            __builtin_amdgcn_global_load_async_to_lds_b64(
                (g_v2_t)(vi2_t*)(void*)gp, (l_v2_t)(vi2_t*)(void*)lp, 0, 0);
        }
        // B: 64 k x 4 gates x 4 n-quads -> 8 async b128 per thread
        #pragma unroll
        for (int i = 0; i < 8; ++i) {
            int id = tid + i * NT_;
            int n4 = id & 3, g = (id >> 2) & 3, kk = id >> 4;
            const float* gp = Wp + (size_t)(kb + kk) * F4_ + g * F_ + cb + n4 * 4;
            float* lp = BsB + g * B_G + ((kk >> 1) & 1) * B_SEL
                            + (kk >> 2) * 32 + (kk & 1) * 16 + n4 * 4;
            __builtin_amdgcn_global_load_async_to_lds_b128(
                (g_v4_t)(vi4_t*)(void*)gp, (l_v4_t)(vi4_t*)(void*)lp, 0, 0);
        }
#else
        #pragma unroll
        for (int i = 0; i < 8; ++i) {
            int id = tid + i * NT_;
            int kc = id & 15, row = id >> 4;
            float4 v = *(const float4*)(Asrc + (size_t)(row0 + row) * F_ + kb + kc * 4);
            v2f vlo; vlo.x = v.x; vlo.y = v.y;
            v2f vhi; vhi.x = v.z; vhi.y = v.w;
            *(v2f*)&AsB[kc * A_KC + row * 2]         = vlo;
            *(v2f*)&AsB[A_SEL + kc * A_KC + row * 2] = vhi;
        }
        #pragma unroll
        for (int i = 0; i < 8; ++i) {
            int id = tid + i * NT_;
            int n4 = id & 3, g = (id >> 2) & 3, kk = id >> 4;
            float4 w = *(const float4*)(Wp + (size_t)(kb + kk) * F4_ + g * F_ + cb + n4 * 4);
            *(float4*)&BsB[g * B_G + ((kk >> 1) & 1) * B_SEL
                           + (kk >> 2) * 32 + (kk & 1) * 16 + n4 * 4] = w;
        }
#endif
    };

    auto compute = [&](const float* As_, const float* Bs_, float asc) {
        #pragma unroll
        for (int kc = 0; kc < 16; ++kc) {
            v2f a = *(const v2f*)&As_[aoff + kc * A_KC];   // ds_load_b64, no cndmask
            a.x *= asc; a.y *= asc;                         // h-side termination mask
            const int kb32 = kc * 32;
            v2f b0; b0.x = Bs_[0 * B_G + boff + kb32]; b0.y = Bs_[0 * B_G + boff + kb32 + 16];
            v2f b1; b1.x = Bs_[1 * B_G + boff + kb32]; b1.y = Bs_[1 * B_G + boff + kb32 + 16];
            v2f b2; b2.x = Bs_[2 * B_G + boff + kb32]; b2.y = Bs_[2 * B_G + boff + kb32 + 16];
            v2f b3; b3.x = Bs_[3 * B_G + boff + kb32]; b3.y = Bs_[3 * B_G + boff + kb32 + 16];
            acc0 = __builtin_amdgcn_wmma_f32_16x16x4_f32(false, a, false, b0, (short)0, acc0, false, false);
            acc1 = __builtin_amdgcn_wmma_f32_16x16x4_f32(false, a, false, b1, (short)0, acc1, false, false);
            acc2 = __builtin_amdgcn_wmma_f32_16x16x4_f32(false, a, false, b2, (short)0, acc2, false, false);
            acc3 = __builtin_amdgcn_wmma_f32_16x16x4_f32(false, a, false, b3, (short)0, acc3, false, false);
        }
    };

#if defined(USE_ASYNC)
    stage(As, Bs, 0);
    for (int t = 0; t < NTILE; ++t) {
        const int cur = t & 1;
        const int k0  = t * KT_;
        if (t + 1 < NTILE) {
            stage(As + (1 - cur) * ABUF, Bs + (1 - cur) * BBUF, k0 + KT_);
            wait_async_le24();   // oldest 24 (= this tile's data) complete
        } else {
            wait_async_0();
        }
        __syncthreads();
        compute(As + cur * ABUF, Bs + cur * BBUF, (k0 >= F_) ? maskf : 1.0f);
        __syncthreads();
    }
#else
    for (int t = 0; t < NTILE; ++t) {
        const int k0 = t * KT_;
        stage(As, Bs, k0);
        __syncthreads();
        compute(As, Bs, (k0 >= F_) ? maskf : 1.0f);
        __syncthreads();
    }
#endif

    // ---- elementwise gate math + state update ----
    const int gcol = cb + nl;
    const float bi = bias[0 * F_ + gcol];
    const float bf = bias[1 * F_ + gcol];
    const float bg = bias[2 * F_ + gcol];
    const float bo = bias[3 * F_ + gcol];

    #pragma unroll
    for (int i = 0; i < 8; ++i) {
        int row = row0 + wv * 16 + i + (lo ? 0 : 8);
        size_t idx = (size_t)row * F_ + gcol;
        float c_old = (term_t[row] != 0) ? 0.0f : c_state[idx];
        float ig = sigmoidf_(acc0[i] + bi);
        float fg = sigmoidf_(acc1[i] + bf);
        float gg = tanhf(acc2[i] + bg);
        float og = sigmoidf_(acc3[i] + bo);
        float nc = fg * c_old + ig * gg;
        float nh = og * tanhf(nc);
        c_state[idx] = nc;
        h_out[idx]   = nh;
        y_t[idx]     = nh;
    }
}

extern "C" void kernel_launch(void* const* d_in, const int* in_sizes, int n_in,
                              void* d_out, int out_size, void* d_ws, size_t ws_size,
                              hipStream_t stream) {
    const float* x    = (const float*)d_in[0];
    const int*   term = (const int*)  d_in[1];
    const float* c0   = (const float*)d_in[2];
    const float* h0   = (const float*)d_in[3];
    const float* Wi   = (const float*)d_in[4];
    const float* Wh   = (const float*)d_in[5];
    const float* bias = (const float*)d_in[6];
    float* out = (float*)d_out;
    float* ws  = (float*)d_ws;

    const size_t BF = (size_t)B_ * F_;
    float* c_state = ws;
    float* hbuf0   = ws + BF;
    float* hbuf1   = ws + 2 * BF;

    (void)hipMemcpyAsync(c_state, c0, BF * sizeof(float), hipMemcpyDeviceToDevice, stream);
    (void)hipMemcpyAsync(hbuf0,   h0, BF * sizeof(float), hipMemcpyDeviceToDevice, stream);

    dim3 grid(64, 2);
    dim3 block(NT_);

    for (int t = 0; t < 256; ++t) {
        const float* h_in = (t & 1) ? hbuf1 : hbuf0;
        float*       h_o  = (t & 1) ? hbuf0 : hbuf1;
        lstm_step_kernel<<<grid, block, 0, stream>>>(
            x + (size_t)t * BF, term + (size_t)t * B_,
            Wi, Wh, bias, h_in, h_o, c_state, out + (size_t)t * BF);
    }

    (void)hipMemcpyAsync(out + 256 * BF,      c_state, BF * sizeof(float), hipMemcpyDeviceToDevice, stream);
    (void)hipMemcpyAsync(out + 256 * BF + BF, hbuf0,   BF * sizeof(float), hipMemcpyDeviceToDevice, stream);
}